// MIL_15178414424101
// MI455X (gfx1250) — compile-verified
//
#include <hip/hip_runtime.h>
#include <hip/hip_bf16.h>

// ---------------- problem constants (from reference) ----------------
#define N1_  4096
#define N2_  32768
#define C_   512
#define N_   (1 + N1_ + N2_)      // 36865
#define STRIDE_ 10
#define K1_  410                  // ceil(4096/10)
#define K2_  3277                 // ceil(32768/10)
#define K_   (1 + K1_ + K2_)      // 3688
#define BIG_ 1000000.0f
#define KCPAD_MAX 3280            // ceil(K2_/16)*16

// d_out flat layout (floats), reference return order
#define XPOOL_OFF   0
#define A_OFF       (K_ * C_)
#define CLUSTER_OFF (A_OFF + K_ * K_)
#define NTYPE_OFF   (CLUSTER_OFF + N_)
#define TREE_OFF    (NTYPE_OFF + K_)
#define FIT_OFF     (TREE_OFF + K_)
#define NEWXY_OFF   (FIT_OFF + N_)

typedef float v2f __attribute__((ext_vector_type(2)));
typedef float v8f __attribute__((ext_vector_type(8)));

// ------------- gfx1250 async global->LDS gather helpers -------------
#if defined(__AMDGCN__) && __has_builtin(__builtin_amdgcn_global_load_async_to_lds_b32)
#define HAS_ASYNC_LDS 1
#else
#define HAS_ASYNC_LDS 0
#endif

// Pointee types matching the builtin prototypes exactly (from clang diag):
//   b32: '__device__ int *'            -> AS(1)/AS(3) int*
//   b64: 'vector_size(2*sizeof(int)) int __device__ *' -> AS pointer to gcc-vector int2
typedef int v2i_vs __attribute__((vector_size(8)));
typedef __attribute__((address_space(1))) int    gbl_i32;
typedef __attribute__((address_space(3))) int    lds_i32;
typedef __attribute__((address_space(1))) v2i_vs gbl_v2i;
typedef __attribute__((address_space(3))) v2i_vs lds_v2i;

__device__ __forceinline__ void stage_b32(const void* g, void* l) {
#if HAS_ASYNC_LDS
  __builtin_amdgcn_global_load_async_to_lds_b32(
      (gbl_i32*)(uintptr_t)g, (lds_i32*)(uint32_t)(uintptr_t)l, 0, 0);
#else
  *(uint32_t*)l = *(const uint32_t*)g;
#endif
}

__device__ __forceinline__ void stage_b64(const void* g, void* l) {
#if HAS_ASYNC_LDS && __has_builtin(__builtin_amdgcn_global_load_async_to_lds_b64)
  __builtin_amdgcn_global_load_async_to_lds_b64(
      (gbl_v2i*)(uintptr_t)g, (lds_v2i*)(uint32_t)(uintptr_t)l, 0, 0);
#else
  *(uint64_t*)l = *(const uint64_t*)g;
#endif
}

__device__ __forceinline__ void wait_async0() {
#if HAS_ASYNC_LDS
#if __has_builtin(__builtin_amdgcn_s_wait_asynccnt)
  __builtin_amdgcn_s_wait_asynccnt(0);
#else
  asm volatile("s_wait_asynccnt 0" ::: "memory");
#endif
#endif
}

// fast clamp+sqrt: v_med3_f32 + raw v_sqrt_f32 (argmin tolerant to 1-ulp)
__device__ __forceinline__ float clamp_sqrt(float x) {
#if defined(__AMDGCN__) && __has_builtin(__builtin_amdgcn_fmed3f) && __has_builtin(__builtin_amdgcn_sqrtf)
  return __builtin_amdgcn_sqrtf(__builtin_amdgcn_fmed3f(x, 0.0f, 3.0e38f));
#else
  return sqrtf(fmaxf(x, 0.0f));
#endif
}

// ---------------- helpers ----------------
__device__ __forceinline__ int cluster_of(int n, const int* __restrict__ c1,
                                          const int* __restrict__ c2) {
  if (n == 0) return 0;
  if (n <= N1_) return c1[n - 1] + 1;
  return c2[n - 1 - N1_] + 1 + K1_;
}

// ---------------- kernels ----------------
__global__ void zero_f_k(float* __restrict__ p, long n) {
  long i = (long)blockIdx.x * blockDim.x + threadIdx.x;
  long stride = (long)gridDim.x * blockDim.x;
  for (; i < n; i += stride) p[i] = 0.0f;
}

__global__ void norms_k(const float* __restrict__ w1, const float* __restrict__ w2,
                        float* __restrict__ invn) {
  __shared__ float sm[256];
  float s1 = 0.f, s2 = 0.f;
  for (int c = threadIdx.x; c < C_; c += 256) {
    float a = w1[c]; s1 += a * a;
    float b = w2[c]; s2 += b * b;
  }
  sm[threadIdx.x] = s1; __syncthreads();
  for (int o = 128; o > 0; o >>= 1) { if (threadIdx.x < o) sm[threadIdx.x] += sm[threadIdx.x + o]; __syncthreads(); }
  if (threadIdx.x == 0) invn[0] = rsqrtf(sm[0]);
  __syncthreads();
  sm[threadIdx.x] = s2; __syncthreads();
  for (int o = 128; o > 0; o >>= 1) { if (threadIdx.x < o) sm[threadIdx.x] += sm[threadIdx.x + o]; __syncthreads(); }
  if (threadIdx.x == 0) invn[1] = rsqrtf(sm[0]);
}

// one wave32 per row; f = tanh(dot(x_row, w) * invnorm)
__global__ void fitness_k(const float* __restrict__ x, const float* __restrict__ w1,
                          const float* __restrict__ w2, const float* __restrict__ invn,
                          float* __restrict__ fit_out) {
  int wave = (int)((blockIdx.x * (long)blockDim.x + threadIdx.x) >> 5);
  int lane = threadIdx.x & 31;
  if (wave >= N1_ + N2_) return;
  const float* w = (wave < N1_) ? w1 : w2;
  const float* row = x + (size_t)(1 + wave) * C_;
  float s = 0.f;
  for (int c = lane; c < C_; c += 32) s += row[c] * w[c];
  for (int o = 16; o > 0; o >>= 1) s += __shfl_xor(s, o, 32);
  if (lane == 0) fit_out[1 + wave] = tanhf(s * invn[wave < N1_ ? 0 : 1]);
}

// stable ascending rank by counting; scatter every STRIDE-th to cent_idx
__global__ void rank_select_k(const float* __restrict__ f, int n, int* __restrict__ cent_idx) {
  __shared__ float tile[256];
  int i = blockIdx.x * blockDim.x + threadIdx.x;
  float fi = (i < n) ? f[i] : 0.f;
  int rank = 0;
  for (int j0 = 0; j0 < n; j0 += 256) {
    int j = j0 + threadIdx.x;
    tile[threadIdx.x] = (j < n) ? f[j] : 0.f;
    __syncthreads();
    int lim = min(256, n - j0);
    for (int jj = 0; jj < lim; ++jj) {
      float fj = tile[jj];
      int jg = j0 + jj;
      rank += ((fj < fi) || (fj == fi && jg < i)) ? 1 : 0;
    }
    __syncthreads();
  }
  if (i < n && (rank % STRIDE_) == 0) cent_idx[rank / STRIDE_] = i;
}

__global__ void parent_k(const int* __restrict__ tree, const int* __restrict__ c1,
                         int* __restrict__ parent) {
  int i = blockIdx.x * blockDim.x + threadIdx.x;
  if (i >= N2_) return;
  parent[i] = c1[tree[1 + N1_ + i] - 1];
}

__global__ void centpar_k(const int* __restrict__ cent2, const int* __restrict__ parent,
                          int* __restrict__ centpar) {
  int i = blockIdx.x * blockDim.x + threadIdx.x;
  if (i >= K2_) return;
  centpar[i] = parent[cent2[i]];
}

// WMMA cluster assignment. K=4 GEMM computes the FULL squared xy-distance:
//   A[M] = [-2cx, -2cy, |c|^2, 1],  B[N] = [px, py, 1, |p|^2]
//   acc  = |c|^2 + |p|^2 - 2 c.p = dxy^2
// Centroid table staged once per block into LDS via async global->LDS gather
// (ASYNCcnt / s_wait_asynccnt); epilogue: sqrt + |cf-pf| + BIG mask + argmin.
__global__ void cluster_wmma_k(const float* __restrict__ xy, const float* __restrict__ fit,
                               const int* __restrict__ cent_idx,
                               const int* __restrict__ parent,   // may be null
                               const int* __restrict__ cent_par, // may be null
                               int Np, int Kc, int* __restrict__ cluster_out) {
  __shared__ float2 s_cxy[KCPAD_MAX];   // (cx, cy)
  __shared__ float2 s_cfp[KCPAD_MAX];   // (cf, bitcast parent)
  const int use_mask = (cent_par != nullptr);
  const int nct  = (Kc + 15) >> 4;
  const int kpad = nct * 16;

  // ---- stage centroid table (async gather to LDS when available) ----
  for (int m = threadIdx.x; m < kpad; m += blockDim.x) {
    int mc = min(m, Kc - 1);
    int ci = cent_idx[mc];
    if (m + (int)blockDim.x < kpad)
      __builtin_prefetch(cent_idx + min(m + (int)blockDim.x, Kc - 1), 0, 1);
    stage_b64(&xy[2 * ci], &s_cxy[m]);         // cx, cy
    stage_b32(&fit[ci], &s_cfp[m].x);          // cf
    if (use_mask) stage_b32(&cent_par[mc], &s_cfp[m].y);   // parent bits
#if !HAS_ASYNC_LDS
    if (!use_mask) s_cfp[m].y = 0.f;
#endif
  }
  wait_async0();
  __syncthreads();

  // ---- per-wave 16-point tile ----
  int wave = (int)((blockIdx.x * (long)blockDim.x + threadIdx.x) >> 5);
  int lane = threadIdx.x & 31;
  int p0 = wave * 16;
  if (p0 >= Np) return;
  int col = lane & 15;
  int hi  = lane >> 4;

  int pidx = p0 + col;                   // Np is a multiple of 16
  float px = xy[pidx * 2 + 0];
  float py = xy[pidx * 2 + 1];
  float pf = fit[pidx];
  float p2 = px * px + py * py;
  int   pp = parent ? parent[pidx] : 0;

  // B fragment (4x16): VGPR0 = row K=0 (px) / K=2 (1);  VGPR1 = K=1 (py) / K=3 (p2)
  v2f b;
  b.x = hi ? 1.0f : px;
  b.y = hi ? p2   : py;

  float best = 3.4e38f;
  int bestM = 0;
  for (int ct = 0; ct < nct; ++ct) {
    float2 cc = s_cxy[ct * 16 + col];    // centroid M = ct*16 + col
    float c2 = cc.x * cc.x + cc.y * cc.y;
    // A fragment (16x4): VGPR0 = K=0 (-2cx) / K=2 (c2); VGPR1 = K=1 (-2cy) / K=3 (1)
    v2f a;
    a.x = (hi == 0) ? (-2.0f * cc.x) : c2;
    a.y = (hi == 0) ? (-2.0f * cc.y) : 1.0f;

    v8f acc = {};
    acc = __builtin_amdgcn_wmma_f32_16x16x4_f32(
        false, a, false, b, (short)0, acc, false, false);

    // D layout: VGPR v on this lane -> (M = ct*16 + v + 8*hi, N = col)
#pragma unroll
    for (int v = 0; v < 8; ++v) {
      int mm = ct * 16 + v + 8 * hi;
      float2 cfp = s_cfp[mm];            // broadcast ds_load_b64
      float val = clamp_sqrt(acc[v]) + fabsf(cfp.x - pf);
      if (use_mask && (__float_as_int(cfp.y) != pp)) val += BIG_;
      if (mm < Kc && val < best) { best = val; bestM = mm; }
    }
  }
  // combine lane l (M%16 in 0..7) with lane l+16 (8..15); first-occurrence tiebreak
  float ob = __shfl_xor(best, 16, 32);
  int   om = __shfl_xor(bestM, 16, 32);
  if (ob < best || (ob == best && om < bestM)) { best = ob; bestM = om; }
  if (hi == 0) cluster_out[pidx] = bestM;
}

// one block per node: scatter x row, xy, counts into zeroed accumulators
__global__ void pool_acc_k(const float* __restrict__ x, const float* __restrict__ xy,
                           const int* __restrict__ c1, const int* __restrict__ c2,
                           float* __restrict__ xpool, float* __restrict__ newxy,
                           float* __restrict__ counts) {
  int n = blockIdx.x;
  int k = cluster_of(n, c1, c2);
  const float* row = x + (size_t)n * C_;
  float* dst = xpool + (size_t)k * C_;
  for (int c = threadIdx.x; c < C_; c += blockDim.x)
    atomicAdd(&dst[c], row[c]);
  if (threadIdx.x == 0) {
    atomicAdd(&counts[k], 1.0f);
    if (n > 0) {
      atomicAdd(&newxy[2 * k + 0], xy[2 * n + 0]);
      atomicAdd(&newxy[2 * k + 1], xy[2 * n + 1]);
    }
  }
}

__global__ void finalize_xpool_k(float* __restrict__ xpool, const float* __restrict__ counts) {
  long i = (long)blockIdx.x * blockDim.x + threadIdx.x;
  if (i >= (long)K_ * C_) return;
  xpool[i] /= fmaxf(counts[i / C_], 1.0f);
}

__global__ void misc_out_k(const int* __restrict__ c1, const int* __restrict__ c2,
                           const int* __restrict__ centpar, const float* __restrict__ counts,
                           float* __restrict__ out) {
  int i = blockIdx.x * blockDim.x + threadIdx.x;
  if (i < N_) out[CLUSTER_OFF + i] = (float)cluster_of(i, c1, c2);
  if (i < K_) {
    out[NTYPE_OFF + i] = (i == 0) ? 0.f : ((i <= K1_) ? 1.f : 2.f);
    out[TREE_OFF + i]  = (i == 0) ? -1.f : ((i <= K1_) ? 0.f : (float)(centpar[i - 1 - K1_] + 1));
    float cnt = fmaxf(counts[i], 1.0f);
    out[NEWXY_OFF + 2 * i + 0] /= cnt;
    out[NEWXY_OFF + 2 * i + 1] /= cnt;
  }
}

__global__ void adjacency_k(const int* __restrict__ ei, int E,
                            const int* __restrict__ c1, const int* __restrict__ c2,
                            float* __restrict__ A) {
  int e = blockIdx.x * blockDim.x + threadIdx.x;
  int total = E + N_;
  if (e >= total) return;
  int r, c;
  if (e < E) { r = ei[e]; c = ei[E + e]; }
  else       { r = e - E; c = r; }
  int cr = cluster_of(r, c1, c2);
  int cc = cluster_of(c, c1, c2);
  atomicAdd(&A[(size_t)cr * K_ + cc], 1.0f);
}

// ---------------- launcher ----------------
extern "C" void kernel_launch(void* const* d_in, const int* in_sizes, int n_in,
                              void* d_out, int out_size, void* d_ws, size_t ws_size,
                              hipStream_t stream) {
  const float* x    = (const float*)d_in[0];
  const int*   ei   = (const int*)d_in[1];
  // d_in[2] = node_type (unused: regenerated analytically)
  const int*   tree = (const int*)d_in[3];
  const float* xy   = (const float*)d_in[4];
  const float* w1   = (const float*)d_in[5];
  const float* w2   = (const float*)d_in[6];
  float* out = (float*)d_out;
  int E = in_sizes[1] / 2;

  // workspace carve-up
  float* invn   = (float*)d_ws;
  float* counts = invn + 2;
  int* cent1    = (int*)(counts + K_);
  int* cent2    = cent1 + K1_;
  int* cluster1 = cent2 + K2_;
  int* cluster2 = cluster1 + N1_;
  int* parent   = cluster2 + N2_;
  int* centpar  = parent + N2_;

  // 1) zero accumulation regions (deterministic under graph replay)
  zero_f_k<<<2048, 256, 0, stream>>>(out, (long)out_size);
  zero_f_k<<<16, 256, 0, stream>>>(counts, (long)K_);

  // 2) weight norms, fitness (writes d_out fitness region; slot 0 already zero)
  norms_k<<<1, 256, 0, stream>>>(w1, w2, invn);
  {
    int waves = N1_ + N2_;
    fitness_k<<<(waves * 32 + 255) / 256, 256, 0, stream>>>(x, w1, w2, invn, out + FIT_OFF);
  }

  // 3) centroid selection via stable rank-by-counting
  rank_select_k<<<N1_ / 256, 256, 0, stream>>>(out + FIT_OFF + 1, N1_, cent1);
  rank_select_k<<<N2_ / 256, 256, 0, stream>>>(out + FIT_OFF + 1 + N1_, N2_, cent2);

  // 4) level-1 WMMA cluster assignment
  cluster_wmma_k<<<(N1_ / 16) * 32 / 256, 256, 0, stream>>>(
      xy + 2 * 1, out + FIT_OFF + 1, cent1, nullptr, nullptr, N1_, K1_, cluster1);

  // 5) parent chain, then level-2 WMMA cluster assignment (with BIG mask)
  parent_k<<<N2_ / 256, 256, 0, stream>>>(tree, cluster1, parent);
  centpar_k<<<(K2_ + 255) / 256, 256, 0, stream>>>(cent2, parent, centpar);
  cluster_wmma_k<<<(N2_ / 16) * 32 / 256, 256, 0, stream>>>(
      xy + 2 * (1 + N1_), out + FIT_OFF + 1 + N1_, cent2, parent, centpar, N2_, K2_, cluster2);

  // 6) pooled features + xy segment means (zero -> atomics -> divide)
  pool_acc_k<<<N_, 256, 0, stream>>>(x, xy, cluster1, cluster2,
                                     out + XPOOL_OFF, out + NEWXY_OFF, counts);
  finalize_xpool_k<<<(int)(((long)K_ * C_ + 255) / 256), 256, 0, stream>>>(out + XPOOL_OFF, counts);
  misc_out_k<<<(N_ + 255) / 256, 256, 0, stream>>>(cluster1, cluster2, centpar, counts, out);

  // 7) pooled adjacency (edges + self loops)
  adjacency_k<<<(E + N_ + 255) / 256, 256, 0, stream>>>(ei, E, cluster1, cluster2, out + A_OFF);
}